// ChannelDropOut_74792560492548
// MI455X (gfx1250) — compile-verified
//
#include <hip/hip_runtime.h>
#include <hip/hip_bf16.h>
#include <stdint.h>

// Problem constants (from reference): B=64, C=512, H=W=32
#define CCH   512
#define TILE  1024                    // H*W floats per (b,c) tile, contiguous
#define NTILES (64 * 512)             // B*C tiles
#define NELEM  (64u * 512u * 32u * 32u) // 33,554,432
#define INV_BHW (1.0f / 65536.0f)     // 1/(B*H*W)

typedef __attribute__((ext_vector_type(2))) float v2f;
typedef __attribute__((ext_vector_type(4))) float v4f;
typedef __attribute__((ext_vector_type(8))) float v8f;

// ---------------------------------------------------------------------------
// Wave32 sum reduction via V_WMMA_F32_16X16X4_F32 with B == ones(4x16):
//   A layout (ISA 7.12.2): VGPR0 lane L<16 -> A[L][0], lane L>=16 -> A[L-16][2]
//   so a = {p, 0} gives D[m][n] = p_m + p_{m+16}  (independent of n).
//   D layout: lane L<16, VGPR r -> D[r][L]; lane L>=16 -> D[r+8][L-16].
//   Sum of the 8 D VGPRs + shfl_xor(16) == sum of all 32 lane partials.
// EXEC must be all-ones at this point (no divergence before the call).
// ---------------------------------------------------------------------------
__device__ __forceinline__ float wave_sum_wmma(float p) {
    v2f a; a[0] = p;    a[1] = 0.0f;
    v2f b; b[0] = 1.0f; b[1] = 1.0f;
    v8f c = {};
    v8f d = __builtin_amdgcn_wmma_f32_16x16x4_f32(
        /*neg_a=*/false, a, /*neg_b=*/false, b,
        /*c_mod=*/(short)0, c, /*reuse_a=*/false, /*reuse_b=*/false);
    float s = ((d[0] + d[1]) + (d[2] + d[3])) + ((d[4] + d[5]) + (d[6] + d[7]));
    s += __shfl_xor(s, 16, 32);
    return s;  // full wave sum, valid in every lane
}

// ---------------------------------------------------------------------------
// Kernel 0: zero the per-channel sum accumulators in workspace.
// ---------------------------------------------------------------------------
__global__ void cd_init_kernel(float* __restrict__ sums) {
    int i = blockIdx.x * blockDim.x + threadIdx.x;
    if (i < CCH) sums[i] = 0.0f;
}

// ---------------------------------------------------------------------------
// Kernel 1: per-channel sum. One 256-thread block per (b,c) tile of 1024
// contiguous floats. float4 load per lane -> WMMA wave reduce -> LDS across
// 8 waves -> one float atomicAdd per tile into sums[c].
// ---------------------------------------------------------------------------
__global__ void cd_reduce_kernel(const float* __restrict__ X,
                                 float* __restrict__ sums) {
    const unsigned tile = blockIdx.x;        // tile = b*C + c
    const unsigned c    = tile & (CCH - 1);  // C is a power of two

    const v4f* p4 = reinterpret_cast<const v4f*>(X + (size_t)tile * TILE);
    v4f v = p4[threadIdx.x];
    float p = (v[0] + v[1]) + (v[2] + v[3]);

    float wsum = wave_sum_wmma(p);           // uniform across the wave

    __shared__ float lds[8];
    const unsigned wave = threadIdx.x >> 5;
    const unsigned lane = threadIdx.x & 31;
    if (lane == 0) lds[wave] = wsum;
    __syncthreads();

    if (threadIdx.x == 0) {
        float t = ((lds[0] + lds[1]) + (lds[2] + lds[3])) +
                  ((lds[4] + lds[5]) + (lds[6] + lds[7]));
        atomicAdd(&sums[c], t);
    }
}

// ---------------------------------------------------------------------------
// Kernel 2: per-channel finalize. gi/reward/new_alpha/new_beta exactly per
// reference; keep_prob ~ Beta(alpha,beta) and mask ~ Bernoulli(keep_prob) via
// a deterministic hash-based sampler (Johnk-style), since JAX PRNG bitstreams
// are not reproducible here.
// ---------------------------------------------------------------------------
__device__ __forceinline__ uint32_t hash_u32(uint32_t x) {
    x ^= x >> 16; x *= 0x7feb352dU;
    x ^= x >> 15; x *= 0x846ca68bU;
    x ^= x >> 16; return x;
}
__device__ __forceinline__ float u01(uint32_t h) {
    return (float)(h >> 8) * (1.0f / 16777216.0f);  // [0,1)
}

__global__ void cd_finalize_kernel(const float* __restrict__ sums,
                                   const float* __restrict__ alpha,
                                   const float* __restrict__ beta,
                                   float* __restrict__ mask_ws,   // ws + 512
                                   float* __restrict__ out_tail)  // d_out + NELEM
{
    int c = blockIdx.x * blockDim.x + threadIdx.x;
    if (c >= CCH) return;

    float gi     = sums[c] * INV_BHW;
    float reward = (gi >= 0.5f) ? 1.0f : 0.0f;
    float a      = alpha[c];
    float b      = beta[c];

    // Deterministic pseudo Beta(a,b) + Bernoulli (seed 42)
    uint32_t h1 = hash_u32((uint32_t)c * 2654435761u ^ 42u);
    uint32_t h2 = hash_u32(h1 ^ 0x9e3779b9u);
    uint32_t h3 = hash_u32(h2 ^ 0x85ebca6bu);
    float u1 = fmaxf(u01(h1), 1e-7f);
    float u2 = fmaxf(u01(h2), 1e-7f);
    float x  = __powf(u1, 1.0f / fmaxf(a, 1e-6f));
    float y  = __powf(u2, 1.0f / fmaxf(b, 1e-6f));
    float keep = x / fmaxf(x + y, 1e-20f);
    float mask = (u01(h3) < keep) ? 1.0f : 0.0f;

    mask_ws[c]            = mask;                    // consumed by scale kernel
    out_tail[c]           = mask;                    // mask_proba (1,C,1,1)
    out_tail[CCH + c]     = a + reward;              // new_alpha
    out_tail[2 * CCH + c] = b + (1.0f - reward);     // new_beta
}

// ---------------------------------------------------------------------------
// Kernel 3: out = X * mask[c]. One block per (b,c) tile; mask load is block-
// uniform (scalarizes). X read should hit the 192MB L2 (filled by kernel 1);
// the output is streamed with non-temporal stores (never re-read).
// ---------------------------------------------------------------------------
__global__ void cd_scale_kernel(const float* __restrict__ X,
                                const float* __restrict__ mask_ws,
                                float* __restrict__ out) {
    const unsigned tile = blockIdx.x;
    const unsigned c    = tile & (CCH - 1);
    const float m = mask_ws[c];

    const size_t base = (size_t)tile * TILE + (size_t)threadIdx.x * 4;
    v4f v = *reinterpret_cast<const v4f*>(X + base);
    v[0] *= m; v[1] *= m; v[2] *= m; v[3] *= m;
    __builtin_nontemporal_store(v, reinterpret_cast<v4f*>(out + base));
}

// ---------------------------------------------------------------------------
// Launch
// ---------------------------------------------------------------------------
extern "C" void kernel_launch(void* const* d_in, const int* in_sizes, int n_in,
                              void* d_out, int out_size, void* d_ws, size_t ws_size,
                              hipStream_t stream) {
    const float* X     = (const float*)d_in[0];
    const float* alpha = (const float*)d_in[1];
    const float* beta  = (const float*)d_in[2];
    // d_in[3] = current_epoch (unused by the computation)

    float* out      = (float*)d_out;
    float* sums     = (float*)d_ws;          // [0,   512) channel sums
    float* mask_ws  = sums + CCH;            // [512, 1024) channel mask
    float* out_tail = out + NELEM;           // mask_proba | new_alpha | new_beta

    cd_init_kernel<<<(CCH + 255) / 256, 256, 0, stream>>>(sums);
    cd_reduce_kernel<<<NTILES, 256, 0, stream>>>(X, sums);
    cd_finalize_kernel<<<(CCH + 255) / 256, 256, 0, stream>>>(
        sums, alpha, beta, mask_ws, out_tail);
    cd_scale_kernel<<<NTILES, 256, 0, stream>>>(X, mask_ws, out);
}